// E3Hamiltonian_20572893348016
// MI455X (gfx1250) — compile-verified
//
#include <hip/hip_runtime.h>
#include <stdint.h>

#ifndef __has_builtin
#define __has_builtin(x) 0
#endif

typedef __attribute__((ext_vector_type(4))) float float4v;
typedef __attribute__((ext_vector_type(4))) int   int4v;
typedef __attribute__((address_space(1))) int4v* global_i4_p;
typedef __attribute__((address_space(3))) int4v* lds_i4_p;

#define PLANE 169u           // nL*nR = 13*13
#define CCH   8u             // C_real
#define CHUNK (CCH * PLANE)  // 1352 floats per batch element
#define GB    8u             // batch elements staged per block
#define LDSF  (GB * CHUNK)   // 10816 floats = 43264 bytes of LDS
#define VEC4S (LDSF / 4u)    // 2704 float4 tiles per block chunk
#define BLOCK 256

// out = (chA +/- chB) * (1/sqrt(2)); channel tables packed as nibbles (s = out channel)
//   srcA[8] = {0,5,3,0,4,7,1,4}
//   srcB[8] = {2,3,5,2,6,1,7,6}
//   negate  = channels {2,3,5,7}
#define SRCA_PACK 0x41740350u
#define SRCB_PACK 0x67162532u
#define NEG_MASK  0xACu

__global__ __launch_bounds__(BLOCK) void spin_proj_kernel(const float* __restrict__ in,
                                                          float* __restrict__ out,
                                                          unsigned int B) {
  __shared__ __align__(16) float lds_in[LDSF];

  const uint32_t tid = threadIdx.x;
  const uint64_t blockFloatBase = (uint64_t)blockIdx.x * (uint64_t)LDSF;
  const uint64_t totalFloats = (uint64_t)B * (uint64_t)CHUNK;

  // ---------------- Phase 1: async DMA global -> LDS, 128b per lane ----------------
#if !__has_builtin(__builtin_amdgcn_global_load_async_to_lds_b128)
  const uint64_t gbase = (uint64_t)(uintptr_t)in + blockFloatBase * 4ull;
  const uint32_t lds_base = (uint32_t)(uintptr_t)&lds_in[0];  // low 32b of generic = LDS offset
#endif
  for (uint32_t v = tid; v < VEC4S; v += BLOCK) {
    const uint64_t gfloat = blockFloatBase + 4ull * (uint64_t)v;
    if (gfloat + 4ull <= totalFloats) {
#if __has_builtin(__builtin_amdgcn_global_load_async_to_lds_b128)
      __builtin_amdgcn_global_load_async_to_lds_b128(
          (global_i4_p)(in + gfloat),
          (lds_i4_p)(&lds_in[4u * v]),
          0, 0);
#else
      const uint32_t goff = v * 16u;             // byte offset within this block's chunk
      const uint32_t loff = lds_base + v * 16u;  // LDS byte address
      asm volatile("global_load_async_to_lds_b128 %0, %1, %2"
                   :: "v"(loff), "v"(goff), "s"(gbase)
                   : "memory");
#endif
    }
  }
#if __has_builtin(__builtin_amdgcn_s_wait_asynccnt)
  __builtin_amdgcn_s_wait_asynccnt(0);
#else
  asm volatile("s_wait_asynccnt 0" ::: "memory");
#endif
  __syncthreads();

  // ---------------- Phase 2: channel mix from LDS, 128b NT stores ----------------
  const float c = 0.70710678118654752440f;  // 1/sqrt(2)
  for (uint32_t v = tid; v < VEC4S; v += BLOCK) {
    const uint64_t gfloat = blockFloatBase + 4ull * (uint64_t)v;
    if (gfloat + 4ull <= totalFloats) {
      float4v r;
#pragma unroll
      for (uint32_t k = 0; k < 4u; ++k) {
        const uint32_t f   = 4u * v + k;       // flat float index within block chunk
        const uint32_t g   = f / CHUNK;        // batch element within chunk
        const uint32_t rem = f - g * CHUNK;
        const uint32_t s   = rem / PLANE;      // output channel 0..7
        const uint32_t idx = rem - s * PLANE;  // pixel 0..168

        const uint32_t sa = (SRCA_PACK >> (4u * s)) & 7u;
        const uint32_t sb = (SRCB_PACK >> (4u * s)) & 7u;
        const float* pl = &lds_in[g * CHUNK + idx];
        const float a = pl[sa * PLANE];
        float       b = pl[sb * PLANE];
        b = ((NEG_MASK >> s) & 1u) ? -b : b;
        r[k] = (a + b) * c;
      }
      __builtin_nontemporal_store(r, (float4v*)(out + gfloat));
    }
  }
}

extern "C" void kernel_launch(void* const* d_in, const int* in_sizes, int n_in,
                              void* d_out, int out_size, void* d_ws, size_t ws_size,
                              hipStream_t stream) {
  const float* in = (const float*)d_in[0];
  float* out = (float*)d_out;
  const unsigned int total = (unsigned int)in_sizes[0];   // B * 8 * 169
  const unsigned int B = total / CHUNK;                   // 65536
  const unsigned int blocks = (B + GB - 1u) / GB;         // 8192
  spin_proj_kernel<<<blocks, BLOCK, 0, stream>>>(in, out, B);
}